// Model_19524921327619
// MI455X (gfx1250) — compile-verified
//
// CDNA5 (gfx1250) implementation of the 3-layer equivariant GNN reference.
// Latency-bound problem (~70 MFLOP, <20MB traffic, fits in L2): fuse edge
// work, fp32 end-to-end, per-irrep linears via V_WMMA_F32_16X16X4_F32 with
// component-major feature layout (unconditional b64 A loads) and LDS-staged
// pre-scaled/padded weights (unconditional ds b64 B loads).
#include <hip/hip_runtime.h>
#include <math.h>

#define GG 128
#define NPG 32
#define NN (GG*NPG)      // 4096 nodes
#define EE (GG*128)      // 16384 edges
#define INV_DENOM (1.0f/1.5f)

// ---- CG table offsets within workspace (floats) ----
enum {
  OFF_CG110 = 0,    // 3*3*1 = 9
  OFF_CG111 = 16,   // 27
  OFF_CG112 = 48,   // 45
  OFF_CG121 = 96,   // 45
  OFF_CG122 = 144,  // 75
  OFF_CG132 = 224,  // 105
  OFF_CG211 = 336,  // 45
  OFF_CG212 = 384,  // 75
  OFF_CG213 = 464,  // 105
  OFF_CG220 = 576,  // 25
  OFF_CG221 = 608,  // 75
  OFF_CG222 = 688,  // 125
  OFF_CG231 = 816,  // 105
  OFF_CG232 = 928,  // 175
  OFF_S0    = 1104, // cg(0,l,l) diagonal sign, l=1..3
  OFF_SHS   = 1108, // SH scales for l=2,3
  CG_TOTAL  = 1152
};

// ---- workspace layout (floats) ----
// Feature blocks are COMPONENT-MAJOR: block base + comp*U + u.
#define OFF_SH   ((size_t)CG_TOTAL)            // E*15 edge spherical harmonics
#define OFF_AGG0 (OFF_SH   + (size_t)EE*15)    // N*9   layer0 aggregates
#define OFF_X1   (OFF_AGG0 + (size_t)NN*9)     // N*96  layer0 out: 0e[0:32] 1o[32+k*8+u] 2e[56+k*8+u]
#define OFF_AGG1 (OFF_X1   + (size_t)NN*96)    // N*728 layer1 agg: 0e@0(U48) 1o@48(U64) 1e@240(U16) 2e@288(U64) 2o@608(U24)
#define OFF_X2   (OFF_AGG1 + (size_t)NN*728)   // N*160 layer1 out: 0e@0(32) 1o@32(3x8) 1e@56(3x8) 2e@80(5x8) 2o@120(5x8)
#define OFF_AGG2 (OFF_X2   + (size_t)NN*160)   // N*64  layer2 agg: 0e[0:48] 0o[48:64]
#define OFF_X3   (OFF_AGG2 + (size_t)NN*64)    // N*8   layer2 out: [0]=0o [1:8]=0e
#define OFF_PRED (OFF_X3   + (size_t)NN*8)     // G*8   pooled predictions
// total ~4.61M floats (~18.5 MB) of d_ws

// ============================================================
// Device-side Clebsch-Gordan generation (replicates reference)
// ============================================================
__device__ double dfact_d(int n) {
  double r = 1.0;
  for (int i = 2; i <= n; ++i) r *= (double)i;
  return r;
}

__device__ double su2_cg_dev(int j1, int m1, int j2, int m2, int j3, int m3) {
  if (m3 != m1 + m2) return 0.0;
  int vmin = -j1 + j2 + m3;
  if (-j1 + m1 > vmin) vmin = -j1 + m1;
  if (0 > vmin) vmin = 0;
  int vmax = j2 + j3 + m1;
  if (j3 - j1 + j2 < vmax) vmax = j3 - j1 + j2;
  if (j3 + m3 < vmax) vmax = j3 + m3;
  if (vmax < vmin) return 0.0;
  double c = sqrt((2.0*j3 + 1.0) * dfact_d(j3 + j1 - j2) * dfact_d(j3 - j1 + j2) *
                  dfact_d(j1 + j2 - j3) / dfact_d(j1 + j2 + j3 + 1));
  c *= sqrt(dfact_d(j3 + m3) * dfact_d(j3 - m3) /
            (dfact_d(j1 + m1) * dfact_d(j1 - m1) * dfact_d(j2 + m2) * dfact_d(j2 - m2)));
  double s = 0.0;
  for (int v = vmin; v <= vmax; ++v) {
    double t = dfact_d(j2 + j3 + m1 - v) * dfact_d(j1 - m1 + v) /
               (dfact_d(v) * dfact_d(j3 - j1 + j2 - v) * dfact_d(j3 + m3 - v) *
                dfact_d(v + j1 - j2 - m3));
    s += (((v + j2 + m2) & 1) ? -t : t);
  }
  return c * s;
}

__device__ void calc_q_dev(int l, double* qr, double* qi) {
  int n = 2*l + 1;
  for (int t = 0; t < n*n; ++t) { qr[t] = 0.0; qi[t] = 0.0; }
  double inv = 1.0 / sqrt(2.0);
  for (int m = -l; m < 0; ++m) {
    qr[(l+m)*n + (l-m)] = inv;     // q[l+m, l+|m|] = 1/sqrt2
    qi[(l+m)*n + (l+m)] = -inv;    // q[l+m, l-|m|] = -1j/sqrt2
  }
  qr[l*n + l] = 1.0;
  for (int m = 1; m <= l; ++m) {
    double sg = (m & 1) ? -1.0 : 1.0;
    qr[(l+m)*n + (l+m)] = sg * inv;
    qi[(l+m)*n + (l-m)] = sg * inv;
  }
  // multiply by (-1j)^l
  double fr, fi;
  switch (l & 3) {
    case 0: fr = 1.0;  fi = 0.0;  break;
    case 1: fr = 0.0;  fi = -1.0; break;
    case 2: fr = -1.0; fi = 0.0;  break;
    default: fr = 0.0; fi = 1.0;  break;
  }
  for (int t = 0; t < n*n; ++t) {
    double r = qr[t], im = qi[t];
    qr[t] = r*fr - im*fi;
    qi[t] = r*fi + im*fr;
  }
}

__device__ void calc_cg_dev(int l1, int l2, int l3, float* out) {
  int n1 = 2*l1+1, n2 = 2*l2+1, n3 = 2*l3+1;
  double q1r[49], q1i[49], q2r[49], q2i[49], q3r[49], q3i[49];
  calc_q_dev(l1, q1r, q1i);
  calc_q_dev(l2, q2r, q2i);
  calc_q_dev(l3, q3r, q3i);
  double su2[343];
  for (int a = 0; a < n1; ++a)
    for (int b = 0; b < n2; ++b)
      for (int cc = 0; cc < n3; ++cc)
        su2[(a*n2 + b)*n3 + cc] = su2_cg_dev(l1, a - l1, l2, b - l2, l3, cc - l3);
  double cr[343], ci[343];
  for (int i = 0; i < n1; ++i)
    for (int j = 0; j < n2; ++j)
      for (int k = 0; k < n3; ++k) {
        double sr = 0.0, si = 0.0;
        for (int a = 0; a < n1; ++a)
          for (int b = 0; b < n2; ++b)
            for (int cc = 0; cc < n3; ++cc) {
              double s = su2[(a*n2 + b)*n3 + cc];
              if (s == 0.0) continue;
              double t1r = q1r[a*n1+i], t1i = q1i[a*n1+i];
              double t2r = q2r[b*n2+j], t2i = q2i[b*n2+j];
              double tr = t1r*t2r - t1i*t2i;
              double ti = t1r*t2i + t1i*t2r;
              double t3r = q3r[cc*n3+k], t3i = -q3i[cc*n3+k]; // conj
              sr += (tr*t3r - ti*t3i) * s;
              si += (tr*t3i + ti*t3r) * s;
            }
        cr[(i*n2 + j)*n3 + k] = sr;
        ci[(i*n2 + j)*n3 + k] = si;
      }
  int tot = n1*n2*n3;
  double mx = 0.0;
  for (int t = 0; t < tot; ++t) { double a = fabs(cr[t]); if (a > mx) mx = a; }
  const double* src = (mx < 1e-7) ? ci : cr;
  double ss = 0.0;
  for (int t = 0; t < tot; ++t) ss += src[t]*src[t];
  double scl = sqrt((2.0*l3 + 1.0) / ss);
  for (int t = 0; t < tot; ++t) out[t] = (float)(src[t] * scl);
}

__global__ void setup_kernel(float* ws) {
  if (blockIdx.x != 0 || threadIdx.x != 0) return;
  calc_cg_dev(1,1,0, ws + OFF_CG110);
  calc_cg_dev(1,1,1, ws + OFF_CG111);
  calc_cg_dev(1,1,2, ws + OFF_CG112);
  calc_cg_dev(1,2,1, ws + OFF_CG121);
  calc_cg_dev(1,2,2, ws + OFF_CG122);
  calc_cg_dev(1,3,2, ws + OFF_CG132);
  calc_cg_dev(2,1,1, ws + OFF_CG211);
  calc_cg_dev(2,1,2, ws + OFF_CG212);
  calc_cg_dev(2,1,3, ws + OFF_CG213);
  calc_cg_dev(2,2,0, ws + OFF_CG220);
  calc_cg_dev(2,2,1, ws + OFF_CG221);
  calc_cg_dev(2,2,2, ws + OFF_CG222);
  calc_cg_dev(2,3,1, ws + OFF_CG231);
  calc_cg_dev(2,3,2, ws + OFF_CG232);
  float tmp[49];
  calc_cg_dev(0,1,1, tmp); ws[OFF_S0+0] = tmp[0];
  calc_cg_dev(0,2,2, tmp); ws[OFF_S0+1] = tmp[0];
  calc_cg_dev(0,3,3, tmp); ws[OFF_S0+2] = tmp[0];
  // SH scales exactly as reference's _sh_scales (fixed probe vector)
  double u0 = 0.1, u1 = -0.4, u2 = 0.7;
  double nrm = sqrt(u0*u0 + u1*u1 + u2*u2);
  u0 /= nrm; u1 /= nrm; u2 /= nrm;
  double y1[3] = { sqrt(3.0)*u1, sqrt(3.0)*u2, sqrt(3.0)*u0 };
  double y2[5], raw[7];
  for (int k = 0; k < 5; ++k) {
    double acc = 0.0;
    for (int i = 0; i < 3; ++i)
      for (int j = 0; j < 3; ++j)
        acc += y1[i]*y1[j]*(double)ws[OFF_CG112 + (i*3 + j)*5 + k];
    raw[k] = acc;
  }
  double nn = 0.0; for (int k = 0; k < 5; ++k) nn += raw[k]*raw[k];
  double s2 = sqrt(5.0) / sqrt(nn);
  ws[OFF_SHS+0] = (float)s2;
  for (int k = 0; k < 5; ++k) y2[k] = s2*raw[k];
  for (int k = 0; k < 7; ++k) {
    double acc = 0.0;
    for (int i = 0; i < 5; ++i)
      for (int j = 0; j < 3; ++j)
        acc += y2[i]*y1[j]*(double)ws[OFF_CG213 + (i*3 + j)*7 + k];
    raw[k] = acc;
  }
  nn = 0.0; for (int k = 0; k < 7; ++k) nn += raw[k]*raw[k];
  ws[OFF_SHS+1] = (float)(sqrt(7.0) / sqrt(nn));
}

// ============================================================
// Utility
// ============================================================
__global__ void zero_kernel(float* p, int n) {
  int i = blockIdx.x*blockDim.x + threadIdx.x;
  if (i < n) p[i] = 0.0f;
}

// Tensor-product block, 8 channels: x is comp-major [n1][8]; result scattered
// into comp-major destination dst[k*stride + u] via fp32 global atomics.
__device__ inline void tp8(const float* x, int n1, const float* y, int n2,
                           const float* C, int n3, float* dst, int stride) {
  for (int u = 0; u < 8; ++u)
    for (int k = 0; k < n3; ++k) {
      float acc = 0.0f;
      for (int i = 0; i < n1; ++i) {
        float xv = x[i*8 + u];
        for (int j = 0; j < n2; ++j)
          acc += xv * y[j] * C[(i*n2 + j)*n3 + k];
      }
      atomicAdd(&dst[k*stride + u], acc);
    }
}

// ============================================================
// Edge kernels
// ============================================================
// Spherical harmonics per edge + layer-0 message scatter (node scalar == 1).
__global__ void sh_edge_kernel(const float* __restrict__ pos,
                               const int* __restrict__ snd,
                               const int* __restrict__ rcv,
                               float* ws) {
  int e = blockIdx.x*blockDim.x + threadIdx.x;
  if (e >= EE) return;
  int s = snd[e], r = rcv[e];
  float vx = pos[r*3+0] - pos[s*3+0];
  float vy = pos[r*3+1] - pos[s*3+1];
  float vz = pos[r*3+2] - pos[s*3+2];
  float inv = rsqrtf(vx*vx + vy*vy + vz*vz);
  float ux = vx*inv, uy = vy*inv, uz = vz*inv;
  const float SQ3 = 1.7320508075688772f;
  float y1[3] = { SQ3*uy, SQ3*uz, SQ3*ux };
  float y2[5], y3[7];
  float s2 = ws[OFF_SHS+0], s3 = ws[OFF_SHS+1];
  for (int k = 0; k < 5; ++k) {
    float acc = 0.0f;
    for (int i = 0; i < 3; ++i)
      for (int j = 0; j < 3; ++j)
        acc += y1[i]*y1[j]*ws[OFF_CG112 + (i*3 + j)*5 + k];
    y2[k] = s2*acc;
  }
  for (int k = 0; k < 7; ++k) {
    float acc = 0.0f;
    for (int i = 0; i < 5; ++i)
      for (int j = 0; j < 3; ++j)
        acc += y2[i]*y1[j]*ws[OFF_CG213 + (i*3 + j)*7 + k];
    y3[k] = s3*acc;
  }
  float* sh = ws + OFF_SH + (size_t)e*15;
  for (int j = 0; j < 3; ++j) sh[j]   = y1[j];
  for (int j = 0; j < 5; ++j) sh[3+j] = y2[j];
  for (int j = 0; j < 7; ++j) sh[8+j] = y3[j];
  // layer-0 aggregation (sender scalar feature == 1, cg(0,l,l) = s0l * I)
  float s01 = ws[OFF_S0+0], s02 = ws[OFF_S0+1];
  float* a0 = ws + OFF_AGG0 + (size_t)r*9;
  atomicAdd(&a0[0], 1.0f);
  for (int j = 0; j < 3; ++j) atomicAdd(&a0[1+j], s01*y1[j]);
  for (int j = 0; j < 5; ++j) atomicAdd(&a0[4+j], s02*y2[j]);
}

// Layer-0 linear (U==1): x1 = 0e[0:32] | 1o[32 + k*8 + v] | 2e[56 + k*8 + v]
__global__ void x1_build_kernel(float* ws,
                                const float* __restrict__ w_lin0e,
                                const float* __restrict__ w_lin1o,
                                const float* __restrict__ w_lin2e,
                                const float* __restrict__ w_sc0e) {
  int n = blockIdx.x*blockDim.x + threadIdx.x;
  if (n >= NN) return;
  const float* a = ws + OFF_AGG0 + (size_t)n*9;
  float* x = ws + OFF_X1 + (size_t)n*96;
  float a0 = a[0]*INV_DENOM;
  for (int v = 0; v < 32; ++v) x[v] = a0*w_lin0e[v] + w_sc0e[v];
  for (int k = 0; k < 3; ++k)
    for (int v = 0; v < 8; ++v)
      x[32 + k*8 + v] = (a[1+k]*INV_DENOM)*w_lin1o[v];
  for (int k = 0; k < 5; ++k)
    for (int v = 0; v < 8; ++v)
      x[56 + k*8 + v] = (a[4+k]*INV_DENOM)*w_lin2e[v];
}

// Layer-1 messages: gather x1[sender], tensor products vs SH, scatter into agg1.
__global__ void l1_edge_kernel(float* ws, const int* __restrict__ snd,
                               const int* __restrict__ rcv) {
  __shared__ float cg[CG_TOTAL];
  for (int t = threadIdx.x; t < CG_TOTAL; t += blockDim.x) cg[t] = ws[t];
  __syncthreads();
  int e = blockIdx.x*blockDim.x + threadIdx.x;
  if (e >= EE) return;
  const float* x = ws + OFF_X1 + (size_t)snd[e]*96;
  const float* sh = ws + OFF_SH + (size_t)e*15;
  float y1[3], y2[5], y3[7];
  for (int j = 0; j < 3; ++j) y1[j] = sh[j];
  for (int j = 0; j < 5; ++j) y2[j] = sh[3+j];
  for (int j = 0; j < 7; ++j) y3[j] = sh[8+j];
  const float* x0e = x;        // 32
  const float* x1o = x + 32;   // [3][8]
  const float* x2e = x + 56;   // [5][8]
  float s01 = cg[OFF_S0+0], s02 = cg[OFF_S0+1];
  float* agg = ws + OFF_AGG1 + (size_t)rcv[e]*728;

  // ---- 0e block @0 (U=48): [node 0e | 1o.Y1->0 | 2e.Y2->0]
  for (int u = 0; u < 32; ++u) atomicAdd(&agg[u], x0e[u]);
  tp8(x1o, 3, y1, 3, cg + OFF_CG110, 1, agg + 32, 48);
  tp8(x2e, 5, y2, 5, cg + OFF_CG220, 1, agg + 40, 48);

  // ---- 1o block @48 (U=64, 3 comps): [node 1o | 0e.Y1 x32 | 1o.Y2 | 2e.Y1 | 2e.Y3]
  float* d1o = agg + 48;
  for (int k = 0; k < 3; ++k)
    for (int u = 0; u < 8; ++u) atomicAdd(&d1o[k*64 + u], x1o[k*8 + u]);
  for (int k = 0; k < 3; ++k)
    for (int u = 0; u < 32; ++u) atomicAdd(&d1o[k*64 + 8 + u], x0e[u]*s01*y1[k]);
  tp8(x1o, 3, y2, 5, cg + OFF_CG121, 3, d1o + 40, 64);
  tp8(x2e, 5, y1, 3, cg + OFF_CG211, 3, d1o + 48, 64);
  tp8(x2e, 5, y3, 7, cg + OFF_CG231, 3, d1o + 56, 64);

  // ---- 1e block @240 (U=16, 3 comps): [1o.Y1 | 2e.Y2]
  float* d1e = agg + 240;
  tp8(x1o, 3, y1, 3, cg + OFF_CG111, 3, d1e, 16);
  tp8(x2e, 5, y2, 5, cg + OFF_CG221, 3, d1e + 8, 16);

  // ---- 2e block @288 (U=64, 5 comps): [node 2e | 0e.Y2 x32 | 1o.Y1 | 1o.Y3 | 2e.Y2]
  float* d2e = agg + 288;
  for (int k = 0; k < 5; ++k)
    for (int u = 0; u < 8; ++u) atomicAdd(&d2e[k*64 + u], x2e[k*8 + u]);
  for (int k = 0; k < 5; ++k)
    for (int u = 0; u < 32; ++u) atomicAdd(&d2e[k*64 + 8 + u], x0e[u]*s02*y2[k]);
  tp8(x1o, 3, y1, 3, cg + OFF_CG112, 5, d2e + 40, 64);
  tp8(x1o, 3, y3, 7, cg + OFF_CG132, 5, d2e + 48, 64);
  tp8(x2e, 5, y2, 5, cg + OFF_CG222, 5, d2e + 56, 64);

  // ---- 2o block @608 (U=24, 5 comps): [1o.Y2 | 2e.Y1 | 2e.Y3]
  float* d2o = agg + 608;
  tp8(x1o, 3, y2, 5, cg + OFF_CG122, 5, d2o, 24);
  tp8(x2e, 5, y1, 3, cg + OFF_CG212, 5, d2o + 8, 24);
  tp8(x2e, 5, y3, 7, cg + OFF_CG232, 5, d2o + 16, 24);
}

// Layer-2 messages: only scalar (l=0) outputs feed the targets.
// agg2: 0e[0:48] = [node 0e | 1o.Y1->0 | 2e.Y2->0], 0o[48:64] = [1e.Y1->0 | 2o.Y2->0]
__global__ void l2_edge_kernel(float* ws, const int* __restrict__ snd,
                               const int* __restrict__ rcv) {
  __shared__ float cg[CG_TOTAL];
  for (int t = threadIdx.x; t < CG_TOTAL; t += blockDim.x) cg[t] = ws[t];
  __syncthreads();
  int e = blockIdx.x*blockDim.x + threadIdx.x;
  if (e >= EE) return;
  const float* sh = ws + OFF_SH + (size_t)e*15;
  float y1[3], y2[5];
  for (int j = 0; j < 3; ++j) y1[j] = sh[j];
  for (int j = 0; j < 5; ++j) y2[j] = sh[3+j];
  const float* x = ws + OFF_X2 + (size_t)snd[e]*160;
  float* agg = ws + OFF_AGG2 + (size_t)rcv[e]*64;
  for (int u = 0; u < 32; ++u) atomicAdd(&agg[u], x[u]);     // node 0e passthrough
  tp8(x + 32,  3, y1, 3, cg + OFF_CG110, 1, agg + 32, 48);   // 1o . Y1 -> 0e
  tp8(x + 80,  5, y2, 5, cg + OFF_CG220, 1, agg + 40, 48);   // 2e . Y2 -> 0e
  tp8(x + 56,  3, y1, 3, cg + OFF_CG110, 1, agg + 48, 16);   // 1e . Y1 -> 0o
  tp8(x + 120, 5, y2, 5, cg + OFF_CG220, 1, agg + 56, 16);   // 2o . Y2 -> 0o
}

// ============================================================
// Generic per-irrep linear via V_WMMA_F32_16X16X4_F32.
// Y[n, v, comp] = sum_u X[n, comp*U + u] * W[u,v] * sA
//              (+ sum_s Xsc[n, comp*Usc + s] * Wsc[s,v] * sB)
// One wave per 16-node tile per component. Weights staged to LDS padded to
// 16 columns with scales folded in -> unconditional b64 loads, no EXEC churn.
// Requires U % 4 == 0 and Usc % 4 == 0 (true for all launches here).
// ============================================================
typedef __attribute__((ext_vector_type(2))) float v2f;
typedef __attribute__((ext_vector_type(8))) float v8f;

__global__ void __launch_bounds__(32)
lin_wmma_kernel(const float* __restrict__ X, int ldx, int U,
                const float* __restrict__ W, float sA,
                const float* __restrict__ Xsc, int ldsc, int Usc,
                const float* __restrict__ Wsc, float sB,
                int Mo, float* __restrict__ Y, int ldy) {
  __shared__ float wl[16 * 128];  // [v][u], v-major, u = [0..U) lin, [U..U+Usc) sc
  int lane = threadIdx.x;
  int Utot = U + Usc;
  for (int t = lane; t < 16*Utot; t += 32) {
    int v = t / Utot, u = t - v*Utot;
    float val = 0.0f;
    if (v < Mo) {
      if (u < U) val = W[u*Mo + v] * sA;
      else if (Wsc) val = Wsc[(u - U)*Mo + v] * sB;
    }
    wl[v*Utot + u] = val;
  }
  __syncthreads();
  int m  = lane & 15;      // A: node row within tile; B/D: output column v
  int kh = lane >> 4;      // K-pair selector (ISA fp32 16x4 A layout)
  int comp = blockIdx.y;
  int row = blockIdx.x*16 + m;
  const float* xrow  = X + (size_t)row*ldx + comp*U;
  const float* wrow  = wl + m*Utot;
  v8f c = {0.f,0.f,0.f,0.f,0.f,0.f,0.f,0.f};
  for (int k0 = 0; k0 < U; k0 += 4) {
    int ka = k0 + 2*kh;
    v2f a = *(const v2f*)(xrow + ka);
    v2f b = *(const v2f*)(wrow + ka);
    c = __builtin_amdgcn_wmma_f32_16x16x4_f32(false, a, false, b, (short)0, c, false, false);
  }
  if (Xsc != nullptr) {
    const float* xs = Xsc + (size_t)row*ldsc + comp*Usc;
    for (int k0 = 0; k0 < Usc; k0 += 4) {
      int ka = k0 + 2*kh;
      v2f a = *(const v2f*)(xs + ka);
      v2f b = *(const v2f*)(wrow + U + ka);
      c = __builtin_amdgcn_wmma_f32_16x16x4_f32(false, a, false, b, (short)0, c, false, false);
    }
  }
  if (m < Mo) {
    for (int r = 0; r < 8; ++r) {
      int mm = blockIdx.x*16 + r + 8*kh;
      Y[(size_t)mm*ldy + comp*Mo + m] = c[r];
    }
  }
}

// ============================================================
// Pool per graph + final output transform
// ============================================================
__global__ void pool_kernel(float* ws, const int* __restrict__ gid) {
  int n = blockIdx.x*blockDim.x + threadIdx.x;
  if (n >= NN) return;
  int g = gid[n];
  const float* x3 = ws + OFF_X3 + (size_t)n*8;
  float* pred = ws + OFF_PRED + (size_t)g*8;
  for (int j = 0; j < 8; ++j) atomicAdd(&pred[j], x3[j]);
}

__global__ void final_kernel(const float* ws, float* out) {
  int g = blockIdx.x*blockDim.x + threadIdx.x;
  if (g >= GG) return;
  const float* p = ws + OFF_PRED + (size_t)g*8;
  float oe = p[0]*p[1];
  out[g*8 + 0] = oe;
  out[g*8 + 1] = -oe;
  for (int j = 0; j < 6; ++j) out[g*8 + 2 + j] = p[2+j];
}

// ============================================================
// Host orchestration
// ============================================================
extern "C" void kernel_launch(void* const* d_in, const int* in_sizes, int n_in,
                              void* d_out, int out_size, void* d_ws, size_t ws_size,
                              hipStream_t stream) {
  (void)in_sizes; (void)n_in; (void)out_size; (void)ws_size;
  // Inputs (pytree order: positions, params sorted by key, senders, receivers, graph_ids)
  const float* pos        = (const float*)d_in[0];
  const float* w_l0_lin0e = (const float*)d_in[1];   // (1,32)
  const float* w_l0_lin1o = (const float*)d_in[2];   // (1,8)
  const float* w_l0_lin2e = (const float*)d_in[3];   // (1,8)
  const float* w_l0_sc0e  = (const float*)d_in[4];   // (1,32)
  const float* w_l1_lin0e = (const float*)d_in[5];   // (48,32)
  const float* w_l1_lin1e = (const float*)d_in[6];   // (16,8)
  const float* w_l1_lin1o = (const float*)d_in[7];   // (64,8)
  const float* w_l1_lin2e = (const float*)d_in[8];   // (64,8)
  const float* w_l1_lin2o = (const float*)d_in[9];   // (24,8)
  const float* w_l1_sc0e  = (const float*)d_in[10];  // (32,32)
  const float* w_l1_sc1o  = (const float*)d_in[11];  // (8,8)
  const float* w_l1_sc2e  = (const float*)d_in[12];  // (8,8)
  const float* w_l2_lin0e = (const float*)d_in[13];  // (48,7)
  const float* w_l2_lin0o = (const float*)d_in[14];  // (16,1)
  const float* w_l2_sc0e  = (const float*)d_in[15];  // (32,7)
  const int* snd = (const int*)d_in[16];
  const int* rcv = (const int*)d_in[17];
  const int* gid = (const int*)d_in[18];
  float* ws  = (float*)d_ws;
  float* out = (float*)d_out;

  // zero the accumulation buffers (graph replay safe: done every call)
  {
    int n0 = NN*9;   zero_kernel<<<(n0+255)/256, 256, 0, stream>>>(ws + OFF_AGG0, n0);
    int n1 = NN*728; zero_kernel<<<(n1+255)/256, 256, 0, stream>>>(ws + OFF_AGG1, n1);
    int n2 = NN*64;  zero_kernel<<<(n2+255)/256, 256, 0, stream>>>(ws + OFF_AGG2, n2);
    int n3 = GG*8;   zero_kernel<<<(n3+255)/256, 256, 0, stream>>>(ws + OFF_PRED, n3);
  }
  setup_kernel<<<1, 1, 0, stream>>>(ws);
  sh_edge_kernel<<<EE/128, 128, 0, stream>>>(pos, snd, rcv, ws);
  x1_build_kernel<<<NN/128, 128, 0, stream>>>(ws, w_l0_lin0e, w_l0_lin1o, w_l0_lin2e, w_l0_sc0e);
  l1_edge_kernel<<<EE/128, 128, 0, stream>>>(ws, snd, rcv);

  const float inv15 = INV_DENOM;
  dim3 wave(32);
  // layer-1 linears (x2: 0e@0 1o@32 1e@56 2e@80 2o@120, comp-major)
  lin_wmma_kernel<<<dim3(NN/16, 1), wave, 0, stream>>>(
      ws + OFF_AGG1, 728, 48, w_l1_lin0e, inv15/sqrtf(48.0f),
      ws + OFF_X1, 96, 32, w_l1_sc0e, 1.0f/sqrtf(32.0f),
      32, ws + OFF_X2, 160);
  lin_wmma_kernel<<<dim3(NN/16, 3), wave, 0, stream>>>(
      ws + OFF_AGG1 + 48, 728, 64, w_l1_lin1o, inv15/sqrtf(64.0f),
      ws + OFF_X1 + 32, 96, 8, w_l1_sc1o, 1.0f/sqrtf(8.0f),
      8, ws + OFF_X2 + 32, 160);
  lin_wmma_kernel<<<dim3(NN/16, 3), wave, 0, stream>>>(
      ws + OFF_AGG1 + 240, 728, 16, w_l1_lin1e, inv15/sqrtf(16.0f),
      nullptr, 0, 0, nullptr, 0.0f,
      8, ws + OFF_X2 + 56, 160);
  lin_wmma_kernel<<<dim3(NN/16, 5), wave, 0, stream>>>(
      ws + OFF_AGG1 + 288, 728, 64, w_l1_lin2e, inv15/sqrtf(64.0f),
      ws + OFF_X1 + 56, 96, 8, w_l1_sc2e, 1.0f/sqrtf(8.0f),
      8, ws + OFF_X2 + 80, 160);
  lin_wmma_kernel<<<dim3(NN/16, 5), wave, 0, stream>>>(
      ws + OFF_AGG1 + 608, 728, 24, w_l1_lin2o, inv15/sqrtf(24.0f),
      nullptr, 0, 0, nullptr, 0.0f,
      8, ws + OFF_X2 + 120, 160);

  l2_edge_kernel<<<EE/128, 128, 0, stream>>>(ws, snd, rcv);
  // layer-2 linears (x3 per node: [0]=0o, [1:8]=0e)
  lin_wmma_kernel<<<dim3(NN/16, 1), wave, 0, stream>>>(
      ws + OFF_AGG2 + 48, 64, 16, w_l2_lin0o, inv15/sqrtf(16.0f),
      nullptr, 0, 0, nullptr, 0.0f,
      1, ws + OFF_X3, 8);
  lin_wmma_kernel<<<dim3(NN/16, 1), wave, 0, stream>>>(
      ws + OFF_AGG2, 64, 48, w_l2_lin0e, inv15/sqrtf(48.0f),
      ws + OFF_X2, 160, 32, w_l2_sc0e, 1.0f/sqrtf(32.0f),
      7, ws + OFF_X3 + 1, 8);

  pool_kernel<<<NN/128, 128, 0, stream>>>(ws, gid);
  final_kernel<<<1, GG, 0, stream>>>(ws, out);
}